// BuildCostVolume_78056735638061
// MI455X (gfx1250) — compile-verified
//
#include <hip/hip_runtime.h>

// BuildCostVolume for MI455X (gfx1250, wave32, WMMA).
// out[b,n,a,k,h,w] = sum_d P[a,k,d] * shear(attn[n])[b,a,d,h,w]
// shear: uh uses in[b,a, d + h - 32, h, w]; vw uses in[b,a, d + w - 32, h, w].

#define DISP_RANGE 10
#define OUTD 21
#define NB 8
#define NA 9
#define ND 128
#define NH 64
#define NW 64
#define PLANE (NH * NW)                  // 4096 floats per depth slice
#define TILES_PER_N (NB * NA * 64 * 4)   // 18432: (b, a, shear-coord, 16-col tile)

typedef __attribute__((ext_vector_type(2))) float v2f;
typedef __attribute__((ext_vector_type(8))) float v8f;

static __device__ __forceinline__ int imin(int x, int y) { return x < y ? x : y; }
static __device__ __forceinline__ int imax(int x, int y) { return x > y ? x : y; }

__global__ __launch_bounds__(256) void BuildCostVolume_78056735638061_kernel(
    const float* __restrict__ uh, const float* __restrict__ vw,
    float* __restrict__ out)
{
    const int lane  = threadIdx.x & 31;
    const int wave  = threadIdx.x >> 5;
    int tile = blockIdx.x * 8 + wave;

    int n = 0;
    if (tile >= TILES_PER_N) { n = 1; tile -= TILES_PER_N; }

    const int ct = tile & 3;           // which 16-wide column tile
    const int rc = (tile >> 2) & 63;   // shear coord: h (uh) or w (vw)
    const int ba = tile >> 8;          // 0..71
    const int a  = ba % NA;
    const int b  = ba / NA;

    const int col   = lane & 15;       // WMMA column (N) this lane owns
    const int khalf = lane >> 4;       // 0: K=+0/+1 ; 1: K=+2/+3 (A/B VGPR layout)

    // ---- pooling-window parameters for this angle a ----
    int ad = a - 4; if (ad < 0) ad = -ad; if (ad == 0) ad = 1;
    const int L      = 20 * ad + 1;
    const int start0 = 64 - 10 * ad;

    // ---- addressing ----
    // uh: elem(d') = d'*PLANE + rc*NW + (ct*16+col), shift = rc-32 (rc==h)
    // vw: elem(d') = d'*PLANE + (ct*16+col)*NW + rc, shift = rc-32 (rc==w)
    const float* __restrict__ src = (n == 0) ? uh : vw;
    const float* __restrict__ p =
        src + (size_t)(b * NA + a) * (ND * PLANE);       // uniform -> SGPR base
    const int colOff = (n == 0) ? (rc * NW + ct * 16 + col)
                                : ((ct * 16 + col) * NW + rc);
    const int shift  = rc - 32;

    // Per-lane A-matrix row band (row m = col; k = kt*16 + col), with the
    // depth-validity condition 0 <= k_idx + shift < ND folded in (loop-invariant).
    int bl0, bh0, bl1, bh1; float v0, v1;
    {
        int k = col;
        int s = (k * L) / OUTD;
        int e = ((k + 1) * L + (OUTD - 1)) / OUTD;
        bl0 = imax(start0 + s, -shift);
        bh0 = imin(start0 + e, ND - shift);
        v0  = 1.0f / (float)(e - s);
        k = 16 + col;
        s = (k * L) / OUTD;
        e = ((k + 1) * L + (OUTD - 1)) / OUTD;
        bl1 = imax(start0 + s, -shift);
        bh1 = imin(start0 + e, ND - shift);
        v1  = (k < OUTD) ? (1.0f / (float)(e - s)) : 0.0f;  // zero rows k>=21
    }

    v8f acc0 = {};   // k = 0..15
    v8f acc1 = {};   // k = 16..20 (+ dead rows)

    const int dd_lo = start0 & ~3;
    const int dd_hi = start0 + L;      // exclusive; P zero outside [start0, start0+L)

    // Branch-free B load: clamp source depth into range; A's tightened band
    // zeroes any contribution whose true depth index was out of bounds.
    auto loadPair = [&](int dd, float& o0, float& o1) {
        const int s0 = dd + 2 * khalf + shift;
        const int c0 = imin(imax(s0, 0), ND - 1);
        const int c1 = imin(imax(s0 + 1, 0), ND - 1);
        o0 = p[(unsigned)(c0 * PLANE + colOff)];
        o1 = p[(unsigned)(c1 * PLANE + colOff)];
    };

    float cb0, cb1;
    loadPair(dd_lo, cb0, cb1);         // prologue

    for (int dd = dd_lo; dd < dd_hi; dd += 4) {
        // Software pipeline: issue next chunk's loads (clamped overfetch is safe)
        float nb0, nb1;
        loadPair(dd + 4, nb0, nb1);

        // Prefetch two chunks ahead along the 16 KB depth stride
        {
            const int sp = imin(imax(dd + 8 + 2 * khalf + shift, 0), ND - 1);
            __builtin_prefetch(&p[(unsigned)(sp * PLANE + colOff)], 0, 1);
        }

        const int kA0 = dd + 2 * khalf;     // K index held in VGPR0 by this lane
        const int kA1 = kA0 + 1;            // K index held in VGPR1

        v2f A0 = { (kA0 >= bl0 && kA0 < bh0) ? v0 : 0.0f,
                   (kA1 >= bl0 && kA1 < bh0) ? v0 : 0.0f };
        v2f A1 = { (kA0 >= bl1 && kA0 < bh1) ? v1 : 0.0f,
                   (kA1 >= bl1 && kA1 < bh1) ? v1 : 0.0f };
        v2f Bv = { cb0, cb1 };

        acc0 = __builtin_amdgcn_wmma_f32_16x16x4_f32(
                   false, A0, false, Bv, (short)0, acc0, false, false);
        acc1 = __builtin_amdgcn_wmma_f32_16x16x4_f32(
                   false, A1, false, Bv, (short)0, acc1, false, false);

        cb0 = nb0; cb1 = nb1;
    }

    // ---- store: D layout (16x16 f32): VGPR r -> row M = r + 8*khalf, col = lane&15
    // out index = outBase + k*PLANE + colOff  (same colOff expression as input tile)
    float* __restrict__ q =
        out + (size_t)((b * 2 + n) * NA + a) * (OUTD * PLANE) + colOff;
#pragma unroll
    for (int r = 0; r < 8; ++r) {
        q[(unsigned)((r + 8 * khalf) * PLANE)] = acc0[r];   // k = 0..15
    }
    // acc1 rows: k = 16 + r + 8*khalf; only khalf==0 lanes hold k=16..20.
    if (khalf == 0) {
#pragma unroll
        for (int r = 0; r < 5; ++r) {
            q[(unsigned)((16 + r) * PLANE)] = acc1[r];
        }
    }
}

extern "C" void kernel_launch(void* const* d_in, const int* in_sizes, int n_in,
                              void* d_out, int out_size, void* d_ws, size_t ws_size,
                              hipStream_t stream) {
    const float* uh = (const float*)d_in[0];
    const float* vw = (const float*)d_in[1];
    float* out = (float*)d_out;

    const int totalWaves = 2 * TILES_PER_N;  // 36864
    dim3 grid(totalWaves / 8);               // 8 waves (256 threads) per block
    dim3 block(256);
    BuildCostVolume_78056735638061_kernel<<<grid, block, 0, stream>>>(uh, vw, out);
}